// Model_6459630813712
// MI455X (gfx1250) — compile-verified
//
#include <hip/hip_runtime.h>
#include <hip/hip_bf16.h>

#define HID 13
#define SEQ 96
#define PRED 720
#define KPAD 16

typedef __attribute__((ext_vector_type(2))) float v2f;
typedef __attribute__((ext_vector_type(8))) float v8f;

// ---------------------------------------------------------------------------
// Kernel 1: 96-step LSTM on the single surviving batch stream (b = B-1).
// One wave32; lane j (<13) owns hidden unit j. Hidden state broadcast via
// wave32 shuffles (no LDS traffic for state). Writes hs padded to (96 x 16)
// into workspace, zeros in columns 13..15 so WMMA K-padding is exact.
// ---------------------------------------------------------------------------
__global__ __launch_bounds__(32) void lstm_stream_kernel(
    const float* __restrict__ x_enc,   // (96, B, 1)
    const float* __restrict__ W_ih,    // (52, 1)
    const float* __restrict__ W_hh,    // (52, 13)
    const float* __restrict__ b_ih,    // (52,)
    const float* __restrict__ b_hh,    // (52,)
    float* __restrict__ hs_pad,        // (96, 16) workspace
    int B)
{
    const int lane    = threadIdx.x;         // 0..31
    const bool active = (lane < HID);
    const int j       = active ? lane : 0;   // clamp so preloads stay in-bounds

    // Register-resident weights: 4 gates (torch order i,f,g,o) x 13 taps.
    float wi[4], bb[4], wh[4][HID];
#pragma unroll
    for (int g = 0; g < 4; ++g) {
        const int row = g * HID + j;
        wi[g] = W_ih[row];
        bb[g] = b_ih[row] + b_hh[row];
#pragma unroll
        for (int k = 0; k < HID; ++k)
            wh[g][k] = W_hh[row * HID + k];
    }

    float h = 0.0f, c = 0.0f;
    const float* xlast = x_enc + (B - 1);    // column b = B-1

    for (int t = 0; t < SEQ; ++t) {
        const float x = xlast[t * B];        // uniform broadcast load

        float s0 = fmaf(x, wi[0], bb[0]);
        float s1 = fmaf(x, wi[1], bb[1]);
        float s2 = fmaf(x, wi[2], bb[2]);
        float s3 = fmaf(x, wi[3], bb[3]);

#pragma unroll
        for (int k = 0; k < HID; ++k) {
            const float hk = __shfl(h, k, 32);   // wave32 broadcast of h[k]
            s0 = fmaf(hk, wh[0][k], s0);
            s1 = fmaf(hk, wh[1][k], s1);
            s2 = fmaf(hk, wh[2][k], s2);
            s3 = fmaf(hk, wh[3][k], s3);
        }

        const float ig = 1.0f / (1.0f + __expf(-s0));
        const float fg = 1.0f / (1.0f + __expf(-s1));
        const float gg = tanhf(s2);
        const float og = 1.0f / (1.0f + __expf(-s3));
        c = fg * c + ig * gg;
        h = og * tanhf(c);

        if (lane < KPAD)
            hs_pad[t * KPAD + lane] = active ? h : 0.0f;  // zero K-padding
    }
}

// ---------------------------------------------------------------------------
// Kernel 1b: pad W_mu (720 x 13) -> w_pad (720 x 16), zeros at K=13..15.
// Same per-row layout as hs_pad so both WMMA fragments load as aligned
// float2 with no divergence (removes the exec-mask branch blocks seen in
// the previous round's codegen).
// ---------------------------------------------------------------------------
__global__ __launch_bounds__(256) void pad_wmu_kernel(
    const float* __restrict__ W_mu,    // (720, 13)
    float* __restrict__ w_pad)         // (720, 16)
{
    const int idx = blockIdx.x * blockDim.x + threadIdx.x;
    if (idx >= PRED * KPAD) return;
    const int n = idx >> 4;
    const int k = idx & 15;
    w_pad[idx] = (k < HID) ? W_mu[n * HID + k] : 0.0f;
}

// ---------------------------------------------------------------------------
// Kernel 2: mu = hs @ W_mu^T + b_mu via fp32 WMMA (V_WMMA_F32_16X16X4_F32).
// Grid = (45 N-tiles, 6 M-tiles), one wave32 per 16x16 output tile.
// K=16 (padded) -> 4 chained K=4 WMMAs; branch-free fragment loads.
//
// 32-bit A 16x4 layout : lanes 0-15 hold {K=0,K=1} of row M=lane,
//                        lanes 16-31 hold {K=2,K=3} of row M=lane-16.
// 32-bit C/D layout    : VGPR r = row (r | half*8), col = lane&15.
// ---------------------------------------------------------------------------
__global__ __launch_bounds__(32) void mu_wmma_kernel(
    const float* __restrict__ hs_pad,  // (96, 16) padded activations
    const float* __restrict__ w_pad,   // (720, 16) padded weights
    const float* __restrict__ b_mu,    // (720,)
    float* __restrict__ out)           // (96, 720, 1)
{
    const int lane = threadIdx.x;
    const int half = lane >> 4;        // 0 = lanes 0-15, 1 = lanes 16-31
    const int l    = lane & 15;
    const int m0   = blockIdx.y * 16;  // 0..80
    const int n    = blockIdx.x * 16 + l;

    const float* arow = hs_pad + (m0 + l) * KPAD + half * 2;
    const float* brow = w_pad  + n * KPAD        + half * 2;

    v8f acc = {};
#pragma unroll
    for (int kk = 0; kk < 4; ++kk) {
        const v2f a = *(const v2f*)(arow + kk * 4);   // 8B-aligned b64 load
        const v2f b = *(const v2f*)(brow + kk * 4);   // 8B-aligned b64 load
        // (neg_a, A, neg_b, B, c_mod, C, reuse_a, reuse_b)
        acc = __builtin_amdgcn_wmma_f32_16x16x4_f32(
            false, a, false, b, (short)0, acc, false, false);
    }

    const float bias = b_mu[n];
#pragma unroll
    for (int r = 0; r < 8; ++r) {
        const int m = m0 + half * 8 + r;
        out[m * PRED + n] = acc[r] + bias;
    }
}

extern "C" void kernel_launch(void* const* d_in, const int* in_sizes, int n_in,
                              void* d_out, int out_size, void* d_ws, size_t ws_size,
                              hipStream_t stream) {
    const float* x_enc = (const float*)d_in[0];
    const float* W_ih  = (const float*)d_in[4];
    const float* W_hh  = (const float*)d_in[5];
    const float* b_ih  = (const float*)d_in[6];
    const float* b_hh  = (const float*)d_in[7];
    const float* W_mu  = (const float*)d_in[8];
    const float* b_mu  = (const float*)d_in[9];
    float* out    = (float*)d_out;
    float* hs_pad = (float*)d_ws;                       // 96*16 floats  (6 KB)
    float* w_pad  = hs_pad + SEQ * KPAD;                // 720*16 floats (45 KB)

    const int B = in_sizes[0] / SEQ;                    // 65536

    lstm_stream_kernel<<<1, 32, 0, stream>>>(x_enc, W_ih, W_hh, b_ih, b_hh,
                                             hs_pad, B);
    pad_wmu_kernel<<<(PRED * KPAD + 255) / 256, 256, 0, stream>>>(W_mu, w_pad);
    mu_wmma_kernel<<<dim3(PRED / 16, SEQ / 16), 32, 0, stream>>>(hs_pad, w_pad,
                                                                 b_mu, out);
}